// SparseMax_19026705121673
// MI455X (gfx1250) — compile-verified
//
#include <hip/hip_runtime.h>

// Sparsemax, N=131072 rows x D=256 fp32.
// HBM-bound: 256 MB traffic -> ~11 us floor @ 23.3 TB/s.
// One wave32 per row (8 floats/lane in VGPRs), Michelot exact-threshold
// iteration, async global->LDS double buffering (CDNA5 ASYNCcnt path).

#define WAVES_PER_BLOCK 8
#define ROWS_PER_WAVE   4
#define DIM             256
#define MAX_ITERS       24

typedef float v4f __attribute__((ext_vector_type(4)));

__device__ __forceinline__ float wave_reduce_sum(float v) {
    v += __shfl_xor(v, 16, 32);
    v += __shfl_xor(v,  8, 32);
    v += __shfl_xor(v,  4, 32);
    v += __shfl_xor(v,  2, 32);
    v += __shfl_xor(v,  1, 32);
    return v;
}

// Copy one 1KB row (256 fp32) from global into an LDS buffer using the
// CDNA5 async-to-LDS path: 32 lanes x 16B per instruction, 2 instructions.
// The immediate offset applies to BOTH the LDS and global address (ISA 10.x),
// so offset:512 handles the second half of the row.
__device__ __forceinline__ void async_copy_row(uint32_t lds_base,
                                               const float* __restrict__ gsrc,
                                               int lane) {
    uint32_t lds = lds_base + (uint32_t)(lane * 16);
    uint64_t g   = (uint64_t)(uintptr_t)gsrc + (uint64_t)(lane * 16);
    asm volatile("global_load_async_to_lds_b128 %0, %1, off"
                 :: "v"(lds), "v"(g) : "memory");
    asm volatile("global_load_async_to_lds_b128 %0, %1, off offset:512"
                 :: "v"(lds), "v"(g) : "memory");
}

__global__ __launch_bounds__(WAVES_PER_BLOCK * 32)
void sparsemax_gfx1250_kernel(const float* __restrict__ x,
                              float* __restrict__ out,
                              int nrows) {
    __shared__ __align__(16) float sbuf[WAVES_PER_BLOCK * 2 * DIM]; // 16 KB

    const int lane = threadIdx.x & 31;
    const int wave = threadIdx.x >> 5;
    const long long wave_g = (long long)blockIdx.x * WAVES_PER_BLOCK + wave;
    const long long row0   = wave_g * ROWS_PER_WAVE;

    float* buf0 = &sbuf[(wave * 2 + 0) * DIM];
    float* buf1 = &sbuf[(wave * 2 + 1) * DIM];
    // Low 32 bits of the flat shared address == LDS group-segment byte offset.
    const uint32_t lb0 = (uint32_t)(uintptr_t)buf0;
    const uint32_t lb1 = (uint32_t)(uintptr_t)buf1;

    // Prologue: prefetch rows 0 and 1 of this wave's batch.
    if (row0 < nrows)     async_copy_row(lb0, x + row0 * DIM, lane);
    if (row0 + 1 < nrows) async_copy_row(lb1, x + (row0 + 1) * DIM, lane);

    for (int i = 0; i < ROWS_PER_WAVE; ++i) {
        const long long r = row0 + i;
        if (r >= nrows) break;

        // In-order ASYNCcnt: <=2 outstanding means row r's two copies landed
        // while row r+1's copies may still be in flight.
        const bool next_inflight = (i + 1 < ROWS_PER_WAVE) && (r + 1 < nrows);
        if (next_inflight) asm volatile("s_wait_asynccnt 0x2" ::: "memory");
        else               asm volatile("s_wait_asynccnt 0x0" ::: "memory");

        const v4f* b4 = (const v4f*)((i & 1) ? buf1 : buf0);
        v4f lo = b4[lane];        // elements lane*4     .. lane*4+3
        v4f hi = b4[lane + 32];   // elements 128+lane*4 .. 128+lane*4+3

        // Drain DS reads into VGPRs before the buffer is overwritten below.
        asm volatile("s_wait_dscnt 0x0" ::: "memory");

        // Refill the just-consumed buffer with row r+2.
        if (i + 2 < ROWS_PER_WAVE && r + 2 < nrows)
            async_copy_row((i & 1) ? lb1 : lb0, x + (r + 2) * DIM, lane);

        float v[8] = {lo.x, lo.y, lo.z, lo.w, hi.x, hi.y, hi.z, hi.w};

        // Michelot: tau <- (sum_{v>tau} v - 1) / |{v>tau}|; fixed point is the
        // exact sparsemax threshold. tau=-inf makes step 1 = (sum-1)/256.
        float tau = -__builtin_inff();
        for (int t = 0; t < MAX_ITERS; ++t) {
            float    ssup = 0.0f;
            unsigned cnt  = 0;
            #pragma unroll
            for (int j = 0; j < 8; ++j) {
                const bool gt = v[j] > tau;
                ssup += gt ? v[j] : 0.0f;
                // Ballot over 32 lanes == 32 row elements: count on SALU.
                cnt += __builtin_popcount(__builtin_amdgcn_ballot_w32(gt));
            }
            ssup = wave_reduce_sum(ssup);                  // wave-uniform
            const float tnew = (ssup - 1.0f) / (float)cnt; // cnt >= 1 always
            if (tnew == tau) break;                        // uniform branch
            tau = tnew;
        }

        v4f olo, ohi;
        olo.x = fmaxf(v[0] - tau, 0.0f);
        olo.y = fmaxf(v[1] - tau, 0.0f);
        olo.z = fmaxf(v[2] - tau, 0.0f);
        olo.w = fmaxf(v[3] - tau, 0.0f);
        ohi.x = fmaxf(v[4] - tau, 0.0f);
        ohi.y = fmaxf(v[5] - tau, 0.0f);
        ohi.z = fmaxf(v[6] - tau, 0.0f);
        ohi.w = fmaxf(v[7] - tau, 0.0f);

        v4f* orow = (v4f*)(out + r * DIM);
        // Streaming output: 128 MB write stream, never re-read -> bypass cache.
        __builtin_nontemporal_store(olo, &orow[lane]);
        __builtin_nontemporal_store(ohi, &orow[lane + 32]);
    }
}

extern "C" void kernel_launch(void* const* d_in, const int* in_sizes, int n_in,
                              void* d_out, int out_size, void* d_ws, size_t ws_size,
                              hipStream_t stream) {
    (void)n_in; (void)d_ws; (void)ws_size; (void)out_size;
    const float* x  = (const float*)d_in[0];
    float* out      = (float*)d_out;
    const int nrows = in_sizes[0] / DIM;                        // 131072
    const int rows_per_block = WAVES_PER_BLOCK * ROWS_PER_WAVE; // 32
    const int blocks = (nrows + rows_per_block - 1) / rows_per_block; // 4096
    sparsemax_gfx1250_kernel<<<blocks, WAVES_PER_BLOCK * 32, 0, stream>>>(x, out, nrows);
}